// AlignLayer_8486855377200
// MI455X (gfx1250) — compile-verified
//
#include <hip/hip_runtime.h>
#include <math.h>

#define D_DIM 128
#define K_OUT 100

typedef float v2f __attribute__((ext_vector_type(2)));
typedef float v8f __attribute__((ext_vector_type(8)));
typedef unsigned int u32x4 __attribute__((ext_vector_type(4)));
typedef int i32x4 __attribute__((ext_vector_type(4)));
typedef int i32x8 __attribute__((ext_vector_type(8)));

#if defined(__has_builtin)
#  if __has_builtin(__builtin_amdgcn_tensor_load_to_lds) && __has_builtin(__builtin_amdgcn_s_wait_tensorcnt)
#    define HAVE_TDM 1
#  endif
#endif
#ifndef HAVE_TDM
#  define HAVE_TDM 0
#endif

// ---------------- small helpers ----------------

__device__ __forceinline__ float half_reduce_max(float v) {
#pragma unroll
    for (int m = 1; m <= 8; m <<= 1) v = fmaxf(v, __shfl_xor(v, m, 32));
    return v; // reduced within each 16-lane half of the wave32
}
__device__ __forceinline__ float half_reduce_sum(float v) {
#pragma unroll
    for (int m = 1; m <= 8; m <<= 1) v += __shfl_xor(v, m, 32);
    return v;
}

#if HAVE_TDM
// LDS aperture flat addresses carry the LDS byte offset in addr[31:0] (ISA 10.2)
__device__ __forceinline__ unsigned lds_byte_offset(const void* p) {
    return (unsigned)(unsigned long long)p;
}

// Issue one TDM 2D tile load (row-major, 4-byte elements) memory -> LDS.
// D# layout per CDNA5 ISA.
__device__ __forceinline__ void tdm_load_2d_f32(const void* gptr, unsigned lds_off,
                                                unsigned dim0, unsigned dim1,
                                                unsigned tile0, unsigned tile1,
                                                unsigned stride0) {
    unsigned long long ga = (unsigned long long)gptr;
    u32x4 g0;
    g0.x = 1u;                                               // count=1, user descriptor
    g0.y = lds_off;                                          // lds_addr
    g0.z = (unsigned)ga;                                     // global_addr[31:0]
    g0.w = (unsigned)((ga >> 32) & 0x01FFFFFFu) | (2u << 30); // global_addr[56:32] | type=2
    i32x8 g1;
    g1[0] = 0x00020000;                                      // data_size=2 (4 bytes), no flags
    g1[1] = (int)(dim0 << 16);                               // tensor_dim0[15:0] at bits 63:48
    g1[2] = (int)((dim0 >> 16) | (dim1 << 16));              // tensor_dim0[31:16] | tensor_dim1[15:0]
    g1[3] = (int)((dim1 >> 16) | (tile0 << 16));             // tensor_dim1[31:16] | tile_dim0
    g1[4] = (int)(tile1 & 0xFFFFu);                          // tile_dim1 | tile_dim2=0
    g1[5] = (int)stride0;                                    // tensor_dim0_stride[31:0]
    g1[6] = 0;
    g1[7] = 0;
    i32x4 z4 = {};
    i32x8 z8 = {};
    // clang-23 / therock-10.0 arity: (g0, g1, g2, g3, g4, cpol)
    __builtin_amdgcn_tensor_load_to_lds(g0, g1, z4, z4, z8, 0);
}
#endif

// ---------------- degree / norm ----------------

__global__ void k_init_deg(float* __restrict__ deg, int N) {
    int i = blockIdx.x * blockDim.x + threadIdx.x;
    if (i < N) deg[i] = 1.0f;               // self-loop contributes 1
}

__global__ void k_deg(const long long* __restrict__ eidx, float* __restrict__ deg, int E) {
    int i = blockIdx.x * blockDim.x + threadIdx.x;
    if (i >= E) return;
    int d = (int)eidx[(long long)E + i];    // edge_index[1][i]
    unsafeAtomicAdd(&deg[d], 1.0f);
}

__global__ void k_dinv(float* __restrict__ deg, int N) {
    int i = blockIdx.x * blockDim.x + threadIdx.x;
    if (i < N) {
        float d = deg[i];
        deg[i] = (d > 0.0f) ? rsqrtf(d) : 0.0f;
    }
}

__global__ void k_norm(const long long* __restrict__ eidx, const float* __restrict__ dinv,
                       float* __restrict__ norm, int E, int N) {
    int i = blockIdx.x * blockDim.x + threadIdx.x;
    int tot = E + N;
    if (i >= tot) return;
    if (i < E) {
        int s = (int)eidx[i];
        int d = (int)eidx[(long long)E + i];
        norm[i] = dinv[s] * dinv[d];
    } else {
        float v = dinv[i - E];
        norm[i] = v * v;
    }
}

// ---------------- bias init ----------------

__global__ void k_bias_init(float* __restrict__ buf, const float* __restrict__ b, int total) {
    int i = blockIdx.x * blockDim.x + threadIdx.x;
    if (i < total) buf[i] = b[i & (D_DIM - 1)];
}

// ---------------- WMMA GEMM: Out[N,128] = (relu?)In[N,128] @ W[128,128] ----------------

__global__ __launch_bounds__(256)
void k_gemm(const float* __restrict__ In, const float* __restrict__ W,
            float* __restrict__ Out, int N, int relu_in) {
    __shared__ float sW[D_DIM * D_DIM]; // 64 KB of the 320 KB/WGP LDS

#if HAVE_TDM
    if (threadIdx.x < 32) {             // one wave issues the TDM DMA (EXEC-independent)
        tdm_load_2d_f32(W, lds_byte_offset(sW),
                        D_DIM, D_DIM, D_DIM, D_DIM, D_DIM);
        __builtin_amdgcn_s_wait_tensorcnt(0);
    }
#else
    {
        const float4* Wv = (const float4*)W;
        float4* sWv = (float4*)sW;
#pragma unroll
        for (int i = 0; i < (D_DIM * D_DIM / 4) / 256; ++i)
            sWv[threadIdx.x + i * 256] = Wv[threadIdx.x + i * 256];
    }
#endif
    __syncthreads();

    const int wave = threadIdx.x >> 5;
    const int lane = threadIdx.x & 31;
    const int half = lane >> 4;
    const int lm   = lane & 15;
    const int row0 = (blockIdx.x * 8 + wave) * 16;
    if (row0 >= N) return;                       // uniform per wave

    int arow = row0 + lm;
    if (arow >= N) arow = N - 1;                 // clamp (stores are guarded)
    const float* ap = In + (size_t)arow * D_DIM;

    v8f acc[8];
#pragma unroll
    for (int t = 0; t < 8; ++t) { v8f z = {}; acc[t] = z; }

    for (int k = 0; k < D_DIM; k += 4) {
        const int k0 = k + 2 * half;             // A: lanes0-15 K=k,k+1 ; lanes16-31 K=k+2,k+3
        float2 av = *(const float2*)(ap + k0);
        v2f a;
        a.x = relu_in ? fmaxf(av.x, 0.0f) : av.x;
        a.y = relu_in ? fmaxf(av.y, 0.0f) : av.y;
#pragma unroll
        for (int t = 0; t < 8; ++t) {
            const int n = t * 16 + lm;
            v2f b;
            b.x = sW[k0 * D_DIM + n];
            b.y = sW[(k0 + 1) * D_DIM + n];
            acc[t] = __builtin_amdgcn_wmma_f32_16x16x4_f32(
                false, a, false, b, (short)0, acc[t], false, false);
        }
    }

    // C/D layout: lanes0-15 hold M=r in VGPR r; lanes16-31 hold M=8+r
    const int obase = row0 + 8 * half;
#pragma unroll
    for (int r = 0; r < 8; ++r) {
        const int orow = obase + r;
        if (orow < N) {
            float* op = Out + (size_t)orow * D_DIM + lm;
#pragma unroll
            for (int t = 0; t < 8; ++t) op[t * 16] = acc[t][r];
        }
    }
}

// ---------------- edge aggregation: agg[dst] += h[src] * norm (L2-resident) ----------------

__global__ __launch_bounds__(256)
void k_aggregate(const float* __restrict__ h, const long long* __restrict__ eidx,
                 const float* __restrict__ norm, float* __restrict__ agg, int E, int N) {
    int gw = (blockIdx.x * blockDim.x + threadIdx.x) >> 5;     // one wave = one edge
    gw = __builtin_amdgcn_readfirstlane(gw);                   // scalarize edge metadata loads
    const int lane = threadIdx.x & 31;
    const int tot  = E + N;
    if (gw >= tot) return;

    int s, d;
    if (gw < E) {
        s = (int)eidx[gw];
        d = (int)eidx[(long long)E + gw];
    } else {
        s = gw - E; d = s;                       // self loop
    }
    const float w = norm[gw];

    const float4 v = *(const float4*)(h + (size_t)s * D_DIM + lane * 4);
    float* dp = agg + (size_t)d * D_DIM + lane * 4;
    unsafeAtomicAdd(dp + 0, v.x * w);
    unsafeAtomicAdd(dp + 1, v.y * w);
    unsafeAtomicAdd(dp + 2, v.z * w);
    unsafeAtomicAdd(dp + 3, v.w * w);
}

// ---------------- head: Out = softmax(H @ Wa + ba), fused WMMA + softmax ----------------

__global__ __launch_bounds__(256)
void k_head(const float* __restrict__ H, const float* __restrict__ Wa,
            const float* __restrict__ ba, float* __restrict__ Out, int N) {
    __shared__ float sW[D_DIM * K_OUT];          // 50 KB, natural stride 100

#if HAVE_TDM
    if (threadIdx.x < 32) {                      // one contiguous 12800-dword tile
        tdm_load_2d_f32(Wa, lds_byte_offset(sW),
                        D_DIM * K_OUT, 1, D_DIM * K_OUT, 1, D_DIM * K_OUT);
        __builtin_amdgcn_s_wait_tensorcnt(0);
    }
#else
    {
        const float4* Wv = (const float4*)Wa;
        float4* sWv = (float4*)sW;
        for (int i = threadIdx.x; i < (D_DIM * K_OUT) / 4; i += 256)
            sWv[i] = Wv[i];
    }
#endif
    __syncthreads();

    const int wave = threadIdx.x >> 5;
    const int lane = threadIdx.x & 31;
    const int half = lane >> 4;
    const int lm   = lane & 15;
    const int row0 = (blockIdx.x * 8 + wave) * 16;
    if (row0 >= N) return;

    int arow = row0 + lm;
    if (arow >= N) arow = N - 1;
    const float* ap = H + (size_t)arow * D_DIM;

    v8f acc[7];
#pragma unroll
    for (int t = 0; t < 7; ++t) { v8f z = {}; acc[t] = z; }

    for (int k = 0; k < D_DIM; k += 4) {
        const int k0 = k + 2 * half;
        float2 av = *(const float2*)(ap + k0);
        v2f a; a.x = av.x; a.y = av.y;
#pragma unroll
        for (int t = 0; t < 7; ++t) {
            const int n  = t * 16 + lm;
            const int nc = (n < K_OUT) ? n : (K_OUT - 1);   // clamp, then zero-select pad cols
            v2f b;
            b.x = sW[k0 * K_OUT + nc];
            b.y = sW[(k0 + 1) * K_OUT + nc];
            if (n >= K_OUT) { b.x = 0.0f; b.y = 0.0f; }
            acc[t] = __builtin_amdgcn_wmma_f32_16x16x4_f32(
                false, a, false, b, (short)0, acc[t], false, false);
        }
    }

    // fused bias + row softmax; row M=r lives in lanes0-15 / M=8+r in lanes16-31
    const int obase = row0 + 8 * half;
#pragma unroll
    for (int r = 0; r < 8; ++r) {
        const int orow = obase + r;
        float v[7], ev[7];
        float mx = -INFINITY;
#pragma unroll
        for (int t = 0; t < 7; ++t) {
            const int col = t * 16 + lm;
            const bool valid = (col < K_OUT);
            v[t] = valid ? (acc[t][r] + ba[col]) : -INFINITY;
            mx = fmaxf(mx, v[t]);
        }
        mx = half_reduce_max(mx);
        float s = 0.0f;
#pragma unroll
        for (int t = 0; t < 7; ++t) {
            ev[t] = __expf(v[t] - mx);           // expf(-inf) == 0 for padded cols
            s += ev[t];
        }
        s = half_reduce_sum(s);
        const float inv = 1.0f / s;
        if (orow < N) {
            float* op = Out + (size_t)orow * K_OUT;
#pragma unroll
            for (int t = 0; t < 7; ++t) {
                const int col = t * 16 + lm;
                if (col < K_OUT) op[col] = ev[t] * inv;
            }
        }
    }
}

// ---------------- launcher ----------------

extern "C" void kernel_launch(void* const* d_in, const int* in_sizes, int n_in,
                              void* d_out, int out_size, void* d_ws, size_t ws_size,
                              hipStream_t stream) {
    const float*      x    = (const float*)d_in[0];
    const long long*  eidx = (const long long*)d_in[1];
    const float*      W1   = (const float*)d_in[2];
    const float*      b1   = (const float*)d_in[3];
    const float*      W2   = (const float*)d_in[4];
    const float*      b2   = (const float*)d_in[5];
    const float*      Wa   = (const float*)d_in[6];
    const float*      ba   = (const float*)d_in[7];
    float*            out  = (float*)d_out;

    const int N   = in_sizes[0] / D_DIM;
    const int E   = in_sizes[1] / 2;
    const int tot = E + N;

    // workspace carve-up (all fp32): dinv[N] | norm[E+N] | bufA[N*128] | bufB[N*128]
    float* dinv = (float*)d_ws;
    float* norm = dinv + (((size_t)N + 63) & ~(size_t)63);
    float* bufA = norm + (((size_t)tot + 63) & ~(size_t)63);
    float* bufB = bufA + (size_t)N * D_DIM;

    const int ndTotal = N * D_DIM;
    const int strips  = (N + 15) / 16;
    const int gblocks = (strips + 7) / 8;        // 8 waves (wave32) per 256-thread block

    // normalization coefficients
    k_init_deg<<<(N + 255) / 256, 256, 0, stream>>>(dinv, N);
    k_deg<<<(E + 255) / 256, 256, 0, stream>>>(eidx, dinv, E);
    k_dinv<<<(N + 255) / 256, 256, 0, stream>>>(dinv, N);
    k_norm<<<(tot + 255) / 256, 256, 0, stream>>>(eidx, dinv, norm, E, N);

    // GCN conv 1: h = relu(agg(x @ W1) + b1)   (relu deferred into conv-2 A-load)
    k_gemm<<<gblocks, 256, 0, stream>>>(x, W1, bufA, N, 0);
    k_bias_init<<<(ndTotal + 255) / 256, 256, 0, stream>>>(bufB, b1, ndTotal);
    k_aggregate<<<(tot + 7) / 8, 256, 0, stream>>>(bufA, eidx, norm, bufB, E, N);

    // GCN conv 2: h2 = agg(relu(h) @ W2) + b2
    k_gemm<<<gblocks, 256, 0, stream>>>(bufB, W2, bufA, N, 1);
    k_bias_init<<<(ndTotal + 255) / 256, 256, 0, stream>>>(bufB, b2, ndTotal);
    k_aggregate<<<(tot + 7) / 8, 256, 0, stream>>>(bufA, eidx, norm, bufB, E, N);

    // head: softmax(h2 @ Wa + ba)
    k_head<<<gblocks, 256, 0, stream>>>(bufB, Wa, ba, out, N);
}